// IndividualLandmarkNetModified_28509992911089
// MI455X (gfx1250) — compile-verified
//
#include <hip/hip_runtime.h>
#include <hip/hip_bf16.h>

typedef float v2f __attribute__((ext_vector_type(2)));
typedef float v8f __attribute__((ext_vector_type(8)));

// Problem constants
constexpr int B  = 32;
constexpr int C  = 3072;
constexpr int HW = 784;          // 28*28
constexpr int K  = 9;            // landmarks+bg
constexpr int KP = 16;           // padded N
constexpr int NC = 2000;

// ---------------------------------------------------------------------------
// Kernel 0a: landmarks [9][3072] -> lmT [3072][16] fp32, zero-padded cols 9..15
// ---------------------------------------------------------------------------
__global__ __launch_bounds__(256) void k_lmT(const float* __restrict__ lm,
                                             float* __restrict__ lmT) {
    int idx = blockIdx.x * 256 + threadIdx.x;   // 0 .. 3072*16-1
    int c = idx >> 4, n = idx & 15;
    lmT[idx] = (n < K) ? lm[n * C + c] : 0.0f;
}

// ---------------------------------------------------------------------------
// Kernel 0b: lsq[k] = sum_c lm[k][c]^2   (one block per k)
// ---------------------------------------------------------------------------
__global__ __launch_bounds__(256) void k_lsq(const float* __restrict__ lm,
                                             float* __restrict__ lsq) {
    int k = blockIdx.x;
    float s = 0.0f;
    for (int c = threadIdx.x; c < C; c += 256) {
        float v = lm[k * C + c];
        s = fmaf(v, v, s);
    }
    s += __shfl_xor(s, 16, 32);
    s += __shfl_xor(s, 8, 32);
    s += __shfl_xor(s, 4, 32);
    s += __shfl_xor(s, 2, 32);
    s += __shfl_xor(s, 1, 32);
    __shared__ float sm[8];
    if ((threadIdx.x & 31) == 0) sm[threadIdx.x >> 5] = s;
    __syncthreads();
    if (threadIdx.x == 0) {
        float t = 0.0f;
        for (int i = 0; i < 8; ++i) t += sm[i];
        lsq[k] = t;
    }
}

// ---------------------------------------------------------------------------
// Kernel 1: per (batch, 16-row p-tile) wave:
//   cross[16p][16k] via 768 x WMMA f32 16x16x4 over C, with ||x||^2 fused,
//   then cdist -> softmax over k -> maps (d_out) + mapsT[b][784][16] (ws)
// ---------------------------------------------------------------------------
__global__ __launch_bounds__(256) void k_maps(const float* __restrict__ x,
                                              const float* __restrict__ lmT,
                                              const float* __restrict__ lsq,
                                              float* __restrict__ mapsT,
                                              float* __restrict__ maps_out) {
    const int lane = threadIdx.x & 31;
    const int wave = threadIdx.x >> 5;
    const int tile = blockIdx.x * 8 + wave;      // 0 .. 32*49-1
    const int b    = tile / 49;
    const int p0   = (tile % 49) * 16;
    const int m16  = lane & 15;                  // A: row M (p), B/D: col N (k)
    const int hi   = lane >> 4;
    const int koff = hi * 2;                     // K offset for this half-wave

    const float* xb = x + (size_t)b * C * HW;

    v8f acc = {};
    float sq = 0.0f;
    #pragma unroll 4
    for (int c = 0; c < C; c += 4) {
        v2f a, bv;
        a.x  = xb[(size_t)(c + koff    ) * HW + p0 + m16];   // K = koff
        a.y  = xb[(size_t)(c + koff + 1) * HW + p0 + m16];   // K = koff+1
        bv.x = lmT[(c + koff    ) * KP + m16];
        bv.y = lmT[(c + koff + 1) * KP + m16];
        sq = fmaf(a.x, a.x, sq);
        sq = fmaf(a.y, a.y, sq);
        acc = __builtin_amdgcn_wmma_f32_16x16x4_f32(false, a, false, bv,
                                                    (short)0, acc, false, false);
    }
    // combine half-wave partial squared norms: lanes l and l+16 -> xsq(p0+(l&15))
    float xsq  = sq + __shfl_xor(sq, 16, 32);
    float lsqn = (m16 < K) ? lsq[m16] : 0.0f;

    #pragma unroll
    for (int j = 0; j < 8; ++j) {
        const int Mrow = j + hi * 8;             // D row this lane holds in acc[j]
        float xs    = __shfl(xsq, Mrow, 32);     // xsq for p = p0+Mrow (lane Mrow<16)
        float cross = acc[j];
        float d2    = xs - 2.0f * cross + lsqn;
        float dist  = sqrtf(fmaxf(d2, 0.0f));
        float v     = (m16 < K) ? -dist : -3.0e38f;
        float mx = v;
        mx = fmaxf(mx, __shfl_xor(mx, 1, 32));
        mx = fmaxf(mx, __shfl_xor(mx, 2, 32));
        mx = fmaxf(mx, __shfl_xor(mx, 4, 32));
        mx = fmaxf(mx, __shfl_xor(mx, 8, 32));
        float e = (m16 < K) ? __expf(v - mx) : 0.0f;
        float s = e;
        s += __shfl_xor(s, 1, 32);
        s += __shfl_xor(s, 2, 32);
        s += __shfl_xor(s, 4, 32);
        s += __shfl_xor(s, 8, 32);
        float mp = e / s;
        mapsT[((size_t)b * HW + p0 + Mrow) * KP + m16] = mp;   // padded cols = 0
        if (m16 < K)
            maps_out[((size_t)b * K + m16) * HW + p0 + Mrow] = mp;
    }
}

// ---------------------------------------------------------------------------
// Kernel 2: all_features[b][c][k] = (1/784) * sum_p x[b][c][p]*mapsT[b][p][k]
//   per (batch, 16-row c-tile) wave: 196 x WMMA f32 16x16x4 over p.
//   Also writes modulated+padded copy afPad[b][3072][16] for kernel 3.
// ---------------------------------------------------------------------------
__global__ __launch_bounds__(256) void k_feat(const float* __restrict__ x,
                                              const float* __restrict__ mapsT,
                                              const float* __restrict__ modulation,
                                              float* __restrict__ af_out,
                                              float* __restrict__ afPad) {
    const int lane = threadIdx.x & 31;
    const int wave = threadIdx.x >> 5;
    const int tile = blockIdx.x * 8 + wave;      // 0 .. 32*192-1
    const int b    = tile / 192;
    const int c0   = (tile % 192) * 16;
    const int m16  = lane & 15;
    const int hi   = lane >> 4;
    const int koff = hi * 2;

    const float* xrow = x + (size_t)b * C * HW + (size_t)(c0 + m16) * HW;
    const float* mT   = mapsT + (size_t)b * HW * KP;

    v8f acc = {};
    #pragma unroll 4
    for (int p = 0; p < HW; p += 4) {
        v2f a = *(const v2f*)(xrow + p + koff);              // K = p+koff, p+koff+1
        v2f bv;
        bv.x = mT[(p + koff    ) * KP + m16];
        bv.y = mT[(p + koff + 1) * KP + m16];
        acc = __builtin_amdgcn_wmma_f32_16x16x4_f32(false, a, false, bv,
                                                    (short)0, acc, false, false);
    }
    const float inv = 1.0f / (float)HW;
    #pragma unroll
    for (int j = 0; j < 8; ++j) {
        const int Mrow = j + hi * 8;
        const int c = c0 + Mrow;
        float af  = acc[j] * inv;
        float mod = (m16 < K) ? modulation[c * K + m16] : 0.0f;
        float mv  = (m16 < K) ? af * mod : 0.0f;
        afPad[((size_t)b * C + c) * KP + m16] = mv;
        if (m16 < K)
            af_out[((size_t)b * C + c) * K + m16] = af;
    }
}

// ---------------------------------------------------------------------------
// Kernel 3: scores[b][nc][k] = sum_c fc_w[nc][c] * afPad[b][c][k]
//   per (batch, 16-row nc-tile) wave: 768 x WMMA f32 16x16x4 over C.
// ---------------------------------------------------------------------------
__global__ __launch_bounds__(256) void k_scores(const float* __restrict__ fcw,
                                                const float* __restrict__ afPad,
                                                float* __restrict__ sc_out) {
    const int lane = threadIdx.x & 31;
    const int wave = threadIdx.x >> 5;
    const int tile = blockIdx.x * 8 + wave;      // 0 .. 32*125-1
    const int b    = tile / 125;
    const int nc0  = (tile % 125) * 16;
    const int m16  = lane & 15;
    const int hi   = lane >> 4;
    const int koff = hi * 2;

    const float* wrow = fcw + (size_t)(nc0 + m16) * C;
    const float* afb  = afPad + (size_t)b * C * KP;

    v8f acc = {};
    #pragma unroll 4
    for (int c = 0; c < C; c += 4) {
        v2f a = *(const v2f*)(wrow + c + koff);
        v2f bv;
        bv.x = afb[(c + koff    ) * KP + m16];
        bv.y = afb[(c + koff + 1) * KP + m16];
        acc = __builtin_amdgcn_wmma_f32_16x16x4_f32(false, a, false, bv,
                                                    (short)0, acc, false, false);
    }
    #pragma unroll
    for (int j = 0; j < 8; ++j) {
        const int Mrow = j + hi * 8;
        if (m16 < K)
            sc_out[((size_t)b * NC + nc0 + Mrow) * K + m16] = acc[j];
    }
}

// ---------------------------------------------------------------------------
extern "C" void kernel_launch(void* const* d_in, const int* in_sizes, int n_in,
                              void* d_out, int out_size, void* d_ws, size_t ws_size,
                              hipStream_t stream) {
    const float* x          = (const float*)d_in[0];   // [32,3072,28,28]
    const float* landmarks  = (const float*)d_in[1];   // [9,3072]
    const float* modulation = (const float*)d_in[2];   // [1,3072,9]
    const float* fc_w       = (const float*)d_in[3];   // [2000,3072]

    float* out      = (float*)d_out;
    float* o_af     = out;                                   // 32*3072*9 = 884736
    float* o_maps   = out + (size_t)B * C * K;               // 32*9*784  = 225792
    float* o_scores = o_maps + (size_t)B * K * HW;           // 32*2000*9 = 576000

    float* ws    = (float*)d_ws;
    float* lmT   = ws;                                       // 3072*16   = 49152
    float* lsq   = lmT + (size_t)C * KP;                     // 16
    float* mapsT = lsq + 16;                                 // 32*784*16 = 401408
    float* afPad = mapsT + (size_t)B * HW * KP;              // 32*3072*16 = 1572864

    k_lmT   <<<(C * KP) / 256, 256, 0, stream>>>(landmarks, lmT);
    k_lsq   <<<K, 256, 0, stream>>>(landmarks, lsq);
    k_maps  <<<(B * 49) / 8, 256, 0, stream>>>(x, lmT, lsq, mapsT, o_maps);
    k_feat  <<<(B * (C / 16)) / 8, 256, 0, stream>>>(x, mapsT, modulation, o_af, afPad);
    k_scores<<<(B * (NC / 16)) / 8, 256, 0, stream>>>(fc_w, afPad, o_scores);
}